// state_loss_power_injection_54631984005295
// MI455X (gfx1250) — compile-verified
//
#include <hip/hip_runtime.h>
#include <hip/hip_bf16.h>
#include <math.h>

#define N_NODES 2000
#define N_EDGES 7064
#define NB      8
#define NVEC    16              // 8 batches x {output, labels}
#define YTHR    1e-3f

typedef __attribute__((ext_vector_type(2))) float v2f;
typedef __attribute__((ext_vector_type(8))) float v8f;

// ---------------------------------------------------------------- utilities
__global__ void k_zero_ws(float* p, int n) {
    int i = blockIdx.x * blockDim.x + threadIdx.x;
    if (i < n) p[i] = 0.f;
}

// Pack the 16 complex vectors into planar Vre/Vim [t][n], t=b -> node_output, t=8+b -> node_labels
__global__ void k_pack_vec(const float* __restrict__ node_out,
                           const float* __restrict__ node_lab,
                           float* __restrict__ Vre, float* __restrict__ Vim) {
    int i = blockIdx.x * blockDim.x + threadIdx.x;      // i = b*N_NODES + n
    if (i >= NB * N_NODES) return;
    int b = i / N_NODES, n = i % N_NODES;
    Vre[b * N_NODES + n]        = node_out[2 * i];
    Vim[b * N_NODES + n]        = node_out[2 * i + 1];
    Vre[(8 + b) * N_NODES + n]  = node_lab[2 * i];
    Vim[(8 + b) * N_NODES + n]  = node_lab[2 * i + 1];
}

// Per-batch diagonal deltas: D_b[i] = sum over valid edges (src=i) of (Y_raw[i,j] - b_edge)
__global__ void k_diag_delta(const int* __restrict__ elab, const int* __restrict__ eidx,
                             const float* __restrict__ Y, const float* __restrict__ battr,
                             float* __restrict__ Dre, float* __restrict__ Dim) {
    int idx = blockIdx.x * blockDim.x + threadIdx.x;
    if (idx >= NB * N_EDGES) return;
    int b = idx / N_EDGES, e = idx % N_EDGES;
    if (elab[b * N_EDGES + e] != 0) return;
    int i = eidx[e], j = eidx[N_EDGES + e];
    if (i == j) return;
    size_t off = 2 * ((size_t)i * N_NODES + j);
    atomicAdd(&Dre[b * N_NODES + i], Y[off]     - battr[2 * e]);
    atomicAdd(&Dim[b * N_NODES + i], Y[off + 1] - battr[2 * e + 1]);
}

// ------------------------------------------------- main WMMA GEMM over T(Y_raw)
// OUT[2000 x 16] complex (planar re/im, layout [n*16 + t]) = T(Y_raw) @ Vc_t
// One wave per block. grid = (125 M-tiles, KPARTS K-partitions) = 3125 waves.
#define KPARTS 25
__global__ __launch_bounds__(32)
void k_gemm_wmma(const float* __restrict__ Y,
                 const float* __restrict__ Vre, const float* __restrict__ Vim,
                 float* __restrict__ Ore, float* __restrict__ Oim) {
    const int lane  = threadIdx.x;          // 0..31
    const int m0    = blockIdx.x * 16;      // row tile
    const int kchnk = N_NODES / KPARTS;     // 80
    const int k0    = blockIdx.y * kchnk;
    const int m     = m0 + (lane & 15);     // A rows: M = lane&15 (ISA 32-bit A layout)
    const int khalf = (lane >> 4) * 2;      // lanes 16-31 hold K+2, K+3
    const int t     = lane & 15;            // B/C column = lane&15

    v8f cre = {};
    v8f cim = {};
#pragma unroll 2
    for (int k = k0; k < k0 + kchnk; k += 4) {
        // ---- A operands: thresholded Y_raw real/imag, 2 consecutive complex elems per lane
        const float* yp = Y + 2 * ((size_t)m * N_NODES + (k + khalf));
        float yr0 = yp[0], yi0 = yp[1], yr1 = yp[2], yi1 = yp[3];
        if (fabsf(yr0) < YTHR) { yr0 = 0.f; yi0 = 0.f; }
        if (fabsf(yr1) < YTHR) { yr1 = 0.f; yi1 = 0.f; }
        v2f ar = { yr0, yr1 };
        v2f ai = { yi0, yi1 };
        // ---- B operands (4x16): row K across VGPRs, column N = lane&15
        int kb = k + khalf;
        float br0 = Vre[t * N_NODES + kb], br1 = Vre[t * N_NODES + kb + 1];
        float bi0 = Vim[t * N_NODES + kb], bi1 = Vim[t * N_NODES + kb + 1];
        v2f bre = {  br0,  br1 };
        v2f bim = {  bi0,  bi1 };
        v2f bni = { -bi0, -bi1 };
        // complex MAC via 4 real fp32 WMMAs
        cre = __builtin_amdgcn_wmma_f32_16x16x4_f32(false, ar, false, bre, (short)0, cre, false, false);
        cre = __builtin_amdgcn_wmma_f32_16x16x4_f32(false, ai, false, bni, (short)0, cre, false, false);
        cim = __builtin_amdgcn_wmma_f32_16x16x4_f32(false, ar, false, bim, (short)0, cim, false, false);
        cim = __builtin_amdgcn_wmma_f32_16x16x4_f32(false, ai, false, bre, (short)0, cim, false, false);
    }
    // C/D layout: VGPR r, lanes 0-15 -> M=r, lanes 16-31 -> M=8+r, N=lane&15
    int mr = m0 + (lane >> 4) * 8;
#pragma unroll
    for (int r = 0; r < 8; ++r) {
        atomicAdd(&Ore[(mr + r) * NVEC + t], cre[r]);
        atomicAdd(&Oim[(mr + r) * NVEC + t], cim[r]);
    }
}

// ------------------------------------------------- sparse corrections
// Off-diagonal valid edges: Y[i,j] becomes 0; subtract T(Y_raw[i,j]) * V_t[j] from row i
__global__ void k_edge_corr(const int* __restrict__ elab, const int* __restrict__ eidx,
                            const float* __restrict__ Y,
                            const float* __restrict__ Vre, const float* __restrict__ Vim,
                            float* __restrict__ Ore, float* __restrict__ Oim) {
    int idx = blockIdx.x * blockDim.x + threadIdx.x;
    if (idx >= NB * N_EDGES) return;
    int b = idx / N_EDGES, e = idx % N_EDGES;
    if (elab[b * N_EDGES + e] != 0) return;
    int i = eidx[e], j = eidx[N_EDGES + e];
    if (i == j) return;
    size_t off = 2 * ((size_t)i * N_NODES + j);
    float yr = Y[off], yi = Y[off + 1];
    if (fabsf(yr) < YTHR) return;            // base already thresholded to 0 -> nothing to remove
#pragma unroll
    for (int s = 0; s < 2; ++s) {
        int tt = b + 8 * s;
        float vr = Vre[tt * N_NODES + j], vi = Vim[tt * N_NODES + j];
        atomicAdd(&Ore[i * NVEC + tt], -(yr * vr - yi * vi));
        atomicAdd(&Oim[i * NVEC + tt], -(yr * vi + yi * vr));
    }
}

// Diagonal: replace T(Y_raw[i,i]) contribution with T(Y_raw[i,i] + D_b[i])
__global__ void k_diag_corr(const float* __restrict__ Y,
                            const float* __restrict__ Dre, const float* __restrict__ Dim,
                            const float* __restrict__ Vre, const float* __restrict__ Vim,
                            float* __restrict__ Ore, float* __restrict__ Oim) {
    int idx = blockIdx.x * blockDim.x + threadIdx.x;
    if (idx >= NB * N_NODES) return;
    int b = idx / N_NODES, i = idx % N_NODES;
    size_t off = 2 * ((size_t)i * N_NODES + i);
    float br = Y[off], bi = Y[off + 1];
    float nr = br + Dre[idx], ni = bi + Dim[idx];
    float tor = (fabsf(br) < YTHR) ? 0.f : br;
    float toi = (fabsf(br) < YTHR) ? 0.f : bi;
    float tnr = (fabsf(nr) < YTHR) ? 0.f : nr;
    float tni = (fabsf(nr) < YTHR) ? 0.f : ni;
    float fr = tnr - tor, fi = tni - toi;
    if (fr == 0.f && fi == 0.f) return;
#pragma unroll
    for (int s = 0; s < 2; ++s) {
        int tt = b + 8 * s;
        float vr = Vre[tt * N_NODES + i], vi = Vim[tt * N_NODES + i];
        atomicAdd(&Ore[i * NVEC + tt], fr * vr - fi * vi);
        atomicAdd(&Oim[i * NVEC + tt], fr * vi + fi * vr);
    }
}

// ------------------------------------------------- reductions
__device__ __forceinline__ void block_reduce_add(float val, float* target) {
    __shared__ float s[256];
    int t = threadIdx.x;
    s[t] = val;
    __syncthreads();
#pragma unroll
    for (int o = 128; o > 0; o >>= 1) {
        if (t < o) s[t] += s[t + o];
        __syncthreads();
    }
    if (t == 0) atomicAdd(target, s[0]);
}

__global__ void k_node_loss(const float* __restrict__ no, const float* __restrict__ nl,
                            float* __restrict__ acc) {
    int i = blockIdx.x * blockDim.x + threadIdx.x;
    float v = 0.f;
    if (i < NB * N_NODES * 2) { float d = no[i] - nl[i]; v = d * d; }
    block_reduce_add(v, &acc[0]);
}

__global__ void k_edge_loss(const float* __restrict__ eo, const int* __restrict__ elab,
                            float* __restrict__ acc) {
    int i = blockIdx.x * blockDim.x + threadIdx.x;
    float v = 0.f;
    if (i < NB * N_EDGES) {
        float x0 = eo[2 * i], x1 = eo[2 * i + 1];
        float mx = fmaxf(x0, x1);
        float lse = mx + logf(__expf(x0 - mx) + __expf(x1 - mx));
        float xt = (elab[i] != 0) ? x1 : x0;
        v = lse - xt;                        // -(logp[tgt])
    }
    block_reduce_add(v, &acc[1]);
}

// S = Vc * conj(YV); d = S_out - S_lab; accumulate |d|^2
__global__ void k_pi_loss(const float* __restrict__ Vre, const float* __restrict__ Vim,
                          const float* __restrict__ Ore, const float* __restrict__ Oim,
                          float* __restrict__ acc) {
    int idx = blockIdx.x * blockDim.x + threadIdx.x;
    float v = 0.f;
    if (idx < NB * N_NODES) {
        int b = idx / N_NODES, n = idx % N_NODES;
        // node_output side (t = b)
        float vr = Vre[b * N_NODES + n],        vi = Vim[b * N_NODES + n];
        float yr = Ore[n * NVEC + b],           yi = Oim[n * NVEC + b];
        float sor = vr * yr + vi * yi;
        float soi = vi * yr - vr * yi;
        // node_labels side (t = 8 + b)
        float wr = Vre[(8 + b) * N_NODES + n],  wi = Vim[(8 + b) * N_NODES + n];
        float zr = Ore[n * NVEC + 8 + b],       zi = Oim[n * NVEC + 8 + b];
        float slr = wr * zr + wi * zi;
        float sli = wi * zr - wr * zi;
        float dr = sor - slr, di = soi - sli;
        v = dr * dr + di * di;
    }
    block_reduce_add(v, &acc[2]);
}

__global__ void k_final(const float* __restrict__ acc, float* __restrict__ out) {
    float node_loss = acc[0] / (float)(NB * N_NODES * 2);
    float edge_loss = acc[1] / (float)(NB * N_EDGES);
    float pi_loss   = acc[2] / (float)(NB * N_NODES) * 0.5f;
    out[0] = node_loss + 0.1f * edge_loss + 0.01f * pi_loss;
    out[1] = node_loss;
    out[2] = edge_loss;
    out[3] = pi_loss;
}

// ---------------------------------------------------------------- launcher
extern "C" void kernel_launch(void* const* d_in, const int* in_sizes, int n_in,
                              void* d_out, int out_size, void* d_ws, size_t ws_size,
                              hipStream_t stream) {
    const float* node_out = (const float*)d_in[0];   // [B*N, 2]
    const float* edge_out = (const float*)d_in[1];   // [B*E, 2]
    const float* node_lab = (const float*)d_in[2];   // [B*N, 2]
    const int*   edge_lab = (const int*)  d_in[3];   // [B, E]
    const float* Y_raw    = (const float*)d_in[4];   // [N, N] complex64 -> interleaved f32
    const float* battr    = (const float*)d_in[5];   // [E] complex64 -> interleaved f32
    const int*   eidx     = (const int*)  d_in[6];   // [2, E]
    float* out = (float*)d_out;

    // workspace layout (floats). Zeroed region first.
    float* Ore = (float*)d_ws;            // 2000*16
    float* Oim = Ore + N_NODES * NVEC;    // 2000*16
    float* Dre = Oim + N_NODES * NVEC;    // 8*2000
    float* Dim = Dre + NB * N_NODES;      // 8*2000
    float* acc = Dim + NB * N_NODES;      // 4
    float* Vre = acc + 4;                 // 16*2000
    float* Vim = Vre + NVEC * N_NODES;    // 16*2000
    const int n_zero = 2 * N_NODES * NVEC + 2 * NB * N_NODES + 4;   // 96004

    k_zero_ws  <<<(n_zero + 255) / 256, 256, 0, stream>>>((float*)d_ws, n_zero);
    k_pack_vec <<<(NB * N_NODES + 255) / 256, 256, 0, stream>>>(node_out, node_lab, Vre, Vim);
    k_diag_delta<<<(NB * N_EDGES + 255) / 256, 256, 0, stream>>>(edge_lab, eidx, Y_raw, battr, Dre, Dim);

    dim3 ggrid(N_NODES / 16, KPARTS);     // 125 x 25 single-wave blocks
    k_gemm_wmma<<<ggrid, 32, 0, stream>>>(Y_raw, Vre, Vim, Ore, Oim);

    k_edge_corr<<<(NB * N_EDGES + 255) / 256, 256, 0, stream>>>(edge_lab, eidx, Y_raw, Vre, Vim, Ore, Oim);
    k_diag_corr<<<(NB * N_NODES + 255) / 256, 256, 0, stream>>>(Y_raw, Dre, Dim, Vre, Vim, Ore, Oim);

    k_node_loss<<<(NB * N_NODES * 2 + 255) / 256, 256, 0, stream>>>(node_out, node_lab, acc);
    k_edge_loss<<<(NB * N_EDGES + 255) / 256, 256, 0, stream>>>(edge_out, edge_lab, acc);
    k_pi_loss  <<<(NB * N_NODES + 255) / 256, 256, 0, stream>>>(Vre, Vim, Ore, Oim, acc);
    k_final    <<<1, 1, 0, stream>>>(acc, out);
}